// GNNEncoder_90890097918588
// MI455X (gfx1250) — compile-verified
//
#include <hip/hip_runtime.h>
#include <hip/hip_bf16.h>

typedef __attribute__((ext_vector_type(2))) float v2f;
typedef __attribute__((ext_vector_type(8))) float v8f;

#define NGRAPH 512
#define NPTS   512

__device__ inline v8f wmma4(v2f a, v2f b, v8f c) {
    // D = A(16x4 f32) * B(4x16 f32) + C(16x16 f32)
    return __builtin_amdgcn_wmma_f32_16x16x4_f32(
        false, a, false, b, (short)0, c, false, false);
}

__device__ inline v8f zero8() {
    v8f z;
#pragma unroll
    for (int i = 0; i < 8; ++i) z[i] = 0.0f;
    return z;
}

// Compiler-only memory fence: same-wave DS ops are executed in order by HW,
// we just need to stop the compiler from reordering LDS stores/loads.
__device__ inline void lds_compiler_fence() { asm volatile("" ::: "memory"); }

// ---------------------------------------------------------------------------
// kNN (k=3, includes self). One block per graph. Writes GLOBAL node indices.
// ---------------------------------------------------------------------------
template <int D>
__global__ __launch_bounds__(256) void knn_kernel(
    const float* __restrict__ h, int stride, int* __restrict__ idx) {
    __shared__ float pts[NPTS * D];
    __shared__ float sq[NPTS];
    const int b = blockIdx.x;
    const float* gp = h + (size_t)b * NPTS * stride;

    for (int i = threadIdx.x; i < NPTS; i += 256) {
        float s = 0.0f;
#pragma unroll
        for (int d = 0; d < D; ++d) {
            float v = gp[(size_t)i * stride + d];
            pts[i * D + d] = v;
            s += v * v;
        }
        sq[i] = s;
    }
    __syncthreads();

#pragma unroll
    for (int r = 0; r < 2; ++r) {
        const int i = threadIdx.x * 2 + r;
        float xi[D];
#pragma unroll
        for (int d = 0; d < D; ++d) xi[d] = pts[i * D + d];
        const float sqi = sq[i];

        float d0 = 3.4e38f, d1 = 3.4e38f, d2v = 3.4e38f;
        int i0 = 0, i1 = 0, i2 = 0;
#pragma unroll 4
        for (int j = 0; j < NPTS; ++j) {
            float dot = 0.0f;
#pragma unroll
            for (int d = 0; d < D; ++d) dot += xi[d] * pts[j * D + d];
            const float dist = sqi + sq[j] - 2.0f * dot;
            if (dist < d0) {
                d2v = d1; i2 = i1; d1 = d0; i1 = i0; d0 = dist; i0 = j;
            } else if (dist < d1) {
                d2v = d1; i2 = i1; d1 = dist; i1 = j;
            } else if (dist < d2v) {
                d2v = dist; i2 = j;
            }
        }
        const int base = (b * NPTS + i) * 3;
        idx[base + 0] = b * NPTS + i0;
        idx[base + 1] = b * NPTS + i1;
        idx[base + 2] = b * NPTS + i2;
    }
}

// ---------------------------------------------------------------------------
// Fused EdgeConv MLP:  e=[xi, xj-xi] -> relu(L0) -> relu(L1) -> relu(L2),
// then mean over k=3.  Per wave: 16 nodes = 48 edges = 3 WMMA M-tiles.
//   L0: [16 x K1] x [K1 x 32]   (K1 = 2D padded to mult of 4)
//   L1: [16 x 32] x [32 x 64]
//   L2: [16 x 64] x [64 x 16]   (cols padded to 16, only DOUT valid)
// ---------------------------------------------------------------------------
template <int D, int K1, int DOUT>
__global__ __launch_bounds__(128) void edge_mlp_kernel(
    const float* __restrict__ hin, int stride_in,
    const int* __restrict__ idx,
    const float* __restrict__ W0, const float* __restrict__ B0,
    const float* __restrict__ W1, const float* __restrict__ B1,
    const float* __restrict__ W2, const float* __restrict__ B2,
    float* __restrict__ hout, int stride_out) {
    constexpr int WAVES = 4;
    __shared__ float sW0[K1 * 32];
    __shared__ float sW1[32 * 64];
    __shared__ float sW2[64 * 16];
    __shared__ float sB0[32], sB1[64], sB2[16];
    __shared__ float sStage[WAVES][16 * 64];  // A/D staging (row major)
    __shared__ float sOut[WAVES][48 * 16];    // per-edge final outputs

    const int tid = threadIdx.x;
    // Stage weights (pad unused rows/cols with zero).
    for (int i = tid; i < K1 * 32; i += 128) {
        int r = i / 32, c = i % 32;
        sW0[i] = (r < 2 * D) ? W0[r * 32 + c] : 0.0f;
    }
    for (int i = tid; i < 32 * 64; i += 128) sW1[i] = W1[i];
    for (int i = tid; i < 64 * 16; i += 128) {
        int r = i / 16, c = i % 16;
        sW2[i] = (c < DOUT) ? W2[r * DOUT + c] : 0.0f;
    }
    if (tid < 32) sB0[tid] = B0[tid];
    if (tid < 64) sB1[tid] = B1[tid];
    if (tid < 16) sB2[tid] = (tid < DOUT) ? B2[tid] : 0.0f;
    __syncthreads();

    const int wave = tid >> 5;
    const int lane = tid & 31;
    const int lrow = lane & 15;
    const int lhi  = lane >> 4;  // 0: lanes 0-15, 1: lanes 16-31
    const int node_base = (blockIdx.x * WAVES + wave) * 16;

    float* S = &sStage[wave][0];
    float* O = &sOut[wave][0];

    for (int t = 0; t < 3; ++t) {  // three 16-edge sub-tiles => 48 edges
        const int eb = node_base * 3 + t * 16;

        // -------- stage edge features: S = [16 x K1] row-major ------------
        for (int i = lane; i < 16 * K1; i += 32) {  // 16*K1 is mult of 32
            const int row = i / K1, col = i % K1;
            const int e = eb + row;
            const int n = e / 3;
            const int j = idx[e];
            float v = 0.0f;
            if (col < D) {
                v = hin[(size_t)n * stride_in + col];
            } else if (col < 2 * D) {
                const int c = col - D;
                v = hin[(size_t)j * stride_in + c] -
                    hin[(size_t)n * stride_in + c];
            }
            S[row * K1 + col] = v;
        }
        lds_compiler_fence();

        // -------- Layer 0: out 16 x 32 (two N-tiles) ----------------------
        v8f c0 = zero8(), c1 = zero8();
#pragma unroll
        for (int kk = 0; kk < K1 / 4; ++kk) {
            const int kb = kk * 4 + 2 * lhi;
            v2f a; a.x = S[lrow * K1 + kb]; a.y = S[lrow * K1 + kb + 1];
            v2f bl; bl.x = sW0[kb * 32 + lrow]; bl.y = sW0[(kb + 1) * 32 + lrow];
            c0 = wmma4(a, bl, c0);
            v2f bh; bh.x = sW0[kb * 32 + 16 + lrow]; bh.y = sW0[(kb + 1) * 32 + 16 + lrow];
            c1 = wmma4(a, bh, c1);
        }
        lds_compiler_fence();
#pragma unroll
        for (int r = 0; r < 8; ++r) {  // bias+relu, D-tile -> S = [16 x 32]
            const int row = r + 8 * lhi;
            S[row * 32 + lrow]      = fmaxf(c0[r] + sB0[lrow], 0.0f);
            S[row * 32 + 16 + lrow] = fmaxf(c1[r] + sB0[16 + lrow], 0.0f);
        }
        lds_compiler_fence();

        // -------- Layer 1: out 16 x 64 (four N-tiles), K=32 ---------------
        v8f d[4];
#pragma unroll
        for (int nt = 0; nt < 4; ++nt) d[nt] = zero8();
#pragma unroll
        for (int kk = 0; kk < 8; ++kk) {
            const int kb = kk * 4 + 2 * lhi;
            v2f a; a.x = S[lrow * 32 + kb]; a.y = S[lrow * 32 + kb + 1];
#pragma unroll
            for (int nt = 0; nt < 4; ++nt) {
                v2f bb;
                bb.x = sW1[kb * 64 + nt * 16 + lrow];
                bb.y = sW1[(kb + 1) * 64 + nt * 16 + lrow];
                d[nt] = wmma4(a, bb, d[nt]);
            }
        }
        lds_compiler_fence();
#pragma unroll
        for (int nt = 0; nt < 4; ++nt)
#pragma unroll
            for (int r = 0; r < 8; ++r) {
                const int row = r + 8 * lhi;
                S[row * 64 + nt * 16 + lrow] =
                    fmaxf(d[nt][r] + sB1[nt * 16 + lrow], 0.0f);
            }
        lds_compiler_fence();

        // -------- Layer 2: out 16 x 16 (padded), K=64 ---------------------
        v8f ec = zero8();
#pragma unroll
        for (int kk = 0; kk < 16; ++kk) {
            const int kb = kk * 4 + 2 * lhi;
            v2f a; a.x = S[lrow * 64 + kb]; a.y = S[lrow * 64 + kb + 1];
            v2f bb; bb.x = sW2[kb * 16 + lrow]; bb.y = sW2[(kb + 1) * 16 + lrow];
            ec = wmma4(a, bb, ec);
        }
#pragma unroll
        for (int r = 0; r < 8; ++r) {
            const int row = r + 8 * lhi;
            O[(t * 16 + row) * 16 + lrow] = fmaxf(ec[r] + sB2[lrow], 0.0f);
        }
        lds_compiler_fence();
    }

    __syncthreads();  // also orders sOut stores vs cross-lane loads

    // -------- mean over k=3, write node features --------------------------
    for (int i = lane; i < 16 * DOUT; i += 32) {  // mult of 32
        const int nl = i / DOUT, c = i % DOUT;
        const float m = (O[(3 * nl + 0) * 16 + c] +
                         O[(3 * nl + 1) * 16 + c] +
                         O[(3 * nl + 2) * 16 + c]) * (1.0f / 3.0f);
        hout[(size_t)(node_base + nl) * stride_out + c] = m;
    }
}

// ---------------------------------------------------------------------------
// Global mean pool + final MLP 16 -> 32 (relu) -> 2
// ---------------------------------------------------------------------------
__global__ __launch_bounds__(256) void final_kernel(
    const float* __restrict__ h, const float* __restrict__ W1,
    const float* __restrict__ B1, const float* __restrict__ W2,
    const float* __restrict__ B2, float* __restrict__ out) {
    __shared__ float ps[16][16];
    __shared__ float g[16];
    __shared__ float hid[32];
    const int b = blockIdx.x;
    const int t = threadIdx.x;
    const int c = t & 15, chunk = t >> 4;
    float s = 0.0f;
    for (int n = chunk; n < NPTS; n += 16)
        s += h[((size_t)b * NPTS + n) * 16 + c];
    ps[chunk][c] = s;
    __syncthreads();
    if (t < 16) {
        float s2 = 0.0f;
#pragma unroll
        for (int q = 0; q < 16; ++q) s2 += ps[q][t];
        g[t] = s2 * (1.0f / (float)NPTS);
    }
    __syncthreads();
    if (t < 32) {
        float s3 = B1[t];
#pragma unroll
        for (int i = 0; i < 16; ++i) s3 += g[i] * W1[i * 32 + t];
        hid[t] = fmaxf(s3, 0.0f);
    }
    __syncthreads();
    if (t < 2) {
        float s4 = B2[t];
#pragma unroll
        for (int j = 0; j < 32; ++j) s4 += hid[j] * W2[j * 2 + t];
        out[b * 2 + t] = s4;
    }
}

// ---------------------------------------------------------------------------
extern "C" void kernel_launch(void* const* d_in, const int* in_sizes, int n_in,
                              void* d_out, int out_size, void* d_ws,
                              size_t ws_size, hipStream_t stream) {
    (void)in_sizes; (void)n_in; (void)out_size; (void)ws_size;
    const float* x = (const float*)d_in[0];
    // d_in[1] = batch (equal-sized graphs; unused)
    // params flattened in insertion order: mlp1, mlp2, mlp3, final; (W,b) each
    const float* m1W0 = (const float*)d_in[2];
    const float* m1b0 = (const float*)d_in[3];
    const float* m1W1 = (const float*)d_in[4];
    const float* m1b1 = (const float*)d_in[5];
    const float* m1W2 = (const float*)d_in[6];
    const float* m1b2 = (const float*)d_in[7];
    const float* m2W0 = (const float*)d_in[8];
    const float* m2b0 = (const float*)d_in[9];
    const float* m2W1 = (const float*)d_in[10];
    const float* m2b1 = (const float*)d_in[11];
    const float* m2W2 = (const float*)d_in[12];
    const float* m2b2 = (const float*)d_in[13];
    const float* m3W0 = (const float*)d_in[14];
    const float* m3b0 = (const float*)d_in[15];
    const float* m3W1 = (const float*)d_in[16];
    const float* m3b1 = (const float*)d_in[17];
    const float* m3W2 = (const float*)d_in[18];
    const float* m3b2 = (const float*)d_in[19];
    const float* fW1  = (const float*)d_in[20];
    const float* fb1  = (const float*)d_in[21];
    const float* fW2  = (const float*)d_in[22];
    const float* fb2  = (const float*)d_in[23];

    const size_t nodes = (size_t)NGRAPH * NPTS;  // 262144
    float* hA  = (float*)d_ws;                   // [nodes][16]
    float* hB  = hA + nodes * 16;                // [nodes][16]
    int*   idx = (int*)(hB + nodes * 16);        // [nodes][3]

    const int knn_blocks  = NGRAPH;                 // 512 (one per graph)
    const int edge_blocks = (int)(nodes / 64);      // 4096 (64 nodes/block)

    // EdgeConv 1: d=3, edge feat 6 (pad K to 8), out 6
    knn_kernel<3><<<knn_blocks, 256, 0, stream>>>(x, 3, idx);
    edge_mlp_kernel<3, 8, 6><<<edge_blocks, 128, 0, stream>>>(
        x, 3, idx, m1W0, m1b0, m1W1, m1b1, m1W2, m1b2, hA, 16);

    // EdgeConv 2: d=6, edge feat 12, out 8
    knn_kernel<6><<<knn_blocks, 256, 0, stream>>>(hA, 16, idx);
    edge_mlp_kernel<6, 12, 8><<<edge_blocks, 128, 0, stream>>>(
        hA, 16, idx, m2W0, m2b0, m2W1, m2b1, m2W2, m2b2, hB, 16);

    // EdgeConv 3: d=8, edge feat 16, out 16
    knn_kernel<8><<<knn_blocks, 256, 0, stream>>>(hB, 16, idx);
    edge_mlp_kernel<8, 16, 16><<<edge_blocks, 128, 0, stream>>>(
        hB, 16, idx, m3W0, m3b0, m3W1, m3b1, m3W2, m3b2, hA, 16);

    // Pool + final MLP
    final_kernel<<<NGRAPH, 256, 0, stream>>>(hA, fW1, fb1, fW2, fb2,
                                             (float*)d_out);
}